// MSloss_6296422056512
// MI455X (gfx1250) — compile-verified
//
#include <hip/hip_runtime.h>
#include <math.h>

typedef __attribute__((ext_vector_type(2))) float v2f;
typedef __attribute__((ext_vector_type(8))) float v8f;

#define BETA_F      1e-07f
#define LAMBDA_TV_F 1e-08f

#define B_    4
#define C_OUT 21
#define T_    3
#define HW_   (512*512)
#define SEM_C 256
#define N_    1024      /* 32*32 */

/* workspace layout in floats */
#define OFF_NUM   0                         /* B_*T_*C_OUT = 252 */
#define OFF_SQ    252                       /* 252 */
#define OFF_DEN   504                       /* B_*C_OUT = 84 */
#define OFF_TV    588                       /* 1 */
#define OFF_GRAD  589                       /* 1 */
#define OFF_MAX   592                       /* B_*N_ = 4096 */
#define OFF_S     4688                      /* B_*N_*32 = 131072 */
#define ZERO_CNT  (OFF_S + 131072)          /* everything above is an accumulator */
#define OFF_SEMN  135760                    /* B_*SEM_C*N_ = 1048576 */
#define OFF_FO    (OFF_SEMN + 1048576)      /* B_*C_OUT*N_ = 86016 */
#define OFF_T     (OFF_FO + 86016)          /* B_*N_*32 = 131072 */

#if defined(__has_builtin)
#if __has_builtin(__builtin_amdgcn_global_load_async_to_lds_b128) && \
    __has_builtin(__builtin_amdgcn_s_wait_asynccnt)
#define USE_ASYNC_LDS 1
#endif
#endif

#ifdef USE_ASYNC_LDS
/* match the builtin's parameter types exactly (from hipcc diagnostic):
   arg0: vector_size(16) int, __device__ (AS1) pointer; arg1: AS3 pointer */
typedef int v4i_gcc __attribute__((vector_size(16)));
typedef __attribute__((address_space(1))) v4i_gcc* as1_v4i;
typedef __attribute__((address_space(3))) v4i_gcc* as3_v4i;
#endif

__device__ __forceinline__ float wave_sum(float v) {
#pragma unroll
  for (int off = 16; off; off >>= 1) v += __shfl_down(v, off, 32);
  return v;
}

/* ---------------- zero accumulators ---------------- */
__global__ void zero_accum(float* __restrict__ ws) {
  int i = blockIdx.x * 256 + threadIdx.x;
  if (i < ZERO_CNT) ws[i] = 0.0f;
}

/* ---------------- level-set partial sums ----------------
   grid = (128 chunks, 4 batches), block = 256. target chunk cached in LDS,
   output streamed once (HBM-bound: ~100 MB total). */
__global__ void levelset_partial(const float* __restrict__ outp,
                                 const float* __restrict__ tgt,
                                 float* __restrict__ ws) {
  const int CHUNK = 2048;
  __shared__ float tg[3][CHUNK];
  __shared__ float red[8][8];
  int b = blockIdx.y;
  int base = blockIdx.x * CHUNK;
  int tid = threadIdx.x;
  for (int j = tid; j < CHUNK; j += 256) {
#pragma unroll
    for (int t = 0; t < 3; ++t)
      tg[t][j] = tgt[((size_t)(b * 3 + t)) * HW_ + base + j];
  }
  __syncthreads();
  int lane = tid & 31, wid = tid >> 5;
  for (int c = 0; c < C_OUT; ++c) {
    float a0 = 0.f, a1 = 0.f, a2 = 0.f, a3 = 0.f, a4 = 0.f, a5 = 0.f, a6 = 0.f;
    const float* op = outp + ((size_t)(b * C_OUT + c)) * HW_ + base;
    for (int j = tid; j < CHUNK; j += 256) {
      float o = op[j];
      float t0 = tg[0][j], t1 = tg[1][j], t2 = tg[2][j];
      a0 += o;
      a1 += t0 * o; a2 += t1 * o; a3 += t2 * o;
      a4 += t0 * t0 * o; a5 += t1 * t1 * o; a6 += t2 * t2 * o;
    }
    a0 = wave_sum(a0); a1 = wave_sum(a1); a2 = wave_sum(a2); a3 = wave_sum(a3);
    a4 = wave_sum(a4); a5 = wave_sum(a5); a6 = wave_sum(a6);
    if (lane == 0) {
      red[wid][0] = a0; red[wid][1] = a1; red[wid][2] = a2; red[wid][3] = a3;
      red[wid][4] = a4; red[wid][5] = a5; red[wid][6] = a6;
    }
    __syncthreads();
    if (tid < 7) {
      float s = 0.f;
#pragma unroll
      for (int w = 0; w < 8; ++w) s += red[w][tid];
      if (tid == 0)      atomicAdd(&ws[OFF_DEN + b * C_OUT + c], s);
      else if (tid < 4)  atomicAdd(&ws[OFF_NUM + (b * 3 + (tid - 1)) * C_OUT + c], s);
      else               atomicAdd(&ws[OFF_SQ  + (b * 3 + (tid - 4)) * C_OUT + c], s);
    }
    __syncthreads();
  }
}

/* ---------------- bilinear resize 64x64 -> 32x32 (align_corners) ---------------- */
__global__ void resize_fo(const float* __restrict__ fout, float* __restrict__ ws,
                          const int* __restrict__ flag) {
  if (*flag == 0) return;
  int i = blockIdx.x * 256 + threadIdx.x;
  if (i >= B_ * C_OUT * N_) return;
  int x = i & 31, y = (i >> 5) & 31, bc = i >> 10;
  const float sc = 63.0f / 31.0f;
  float yy = y * sc, xx = x * sc;
  int y0 = (int)floorf(yy); int y1 = (y0 + 1 < 63) ? y0 + 1 : 63;
  int x0 = (int)floorf(xx); int x1 = (x0 + 1 < 63) ? x0 + 1 : 63;
  float wy = yy - (float)y0, wx = xx - (float)x0;
  const float* p = fout + (size_t)bc * 64 * 64;
  float v00 = p[y0 * 64 + x0], v01 = p[y0 * 64 + x1];
  float v10 = p[y1 * 64 + x0], v11 = p[y1 * 64 + x1];
  float top = v00 * (1.f - wx) + v01 * wx;
  float bot = v10 * (1.f - wx) + v11 * wx;
  ws[OFF_FO + i] = top * (1.f - wy) + bot * wy;
}

/* ---------------- channel-L2 normalize f_sem ---------------- */
__global__ void sem_normalize(const float* __restrict__ sem, float* __restrict__ ws,
                              const int* __restrict__ flag) {
  if (*flag == 0) return;
  int i = blockIdx.x * 256 + threadIdx.x;   /* 0..4095 : (b,p) */
  if (i >= B_ * N_) return;
  int b = i >> 10, p = i & 1023;
  const float* sp = sem + (size_t)b * SEM_C * N_ + p;
  float ss = 0.f;
  for (int c = 0; c < SEM_C; ++c) { float v = sp[c * N_]; ss += v * v; }
  float inv = 1.0f / fmaxf(sqrtf(ss), 1e-12f);
  float* dp = ws + OFF_SEMN + (size_t)b * SEM_C * N_ + p;
  for (int c = 0; c < SEM_C; ++c) dp[c * N_] = sp[c * N_] * inv;
}

/* ---------------- tau = relu(sem^T sem) via V_WMMA_F32_16X16X4_F32 ----------------
   64x64 macro-tile per 256-thread block (8 waves x 2 adjacent 16x16 tiles).
   A/B panels staged in LDS via async global->LDS (ASYNCcnt) when available,
   row stride 72 DWORDs so the two 16-lane halves (K vs K+2 rows, 144 DWORDs
   apart, 144 % 64 == 16) hit disjoint bank sets. Fused epilogue: relu,
   per-row max, per-row 32-col-block sums. tau (16 MB dense) never hits memory. */
#define KC 64
#define APAD 72

__global__ void __launch_bounds__(256)
tau_wmma(float* __restrict__ ws, const int* __restrict__ flag) {
  if (*flag == 0) return;
  __shared__ __attribute__((aligned(16))) float As[KC * APAD];
  __shared__ __attribute__((aligned(16))) float Bs[KC * APAD];
  int blk = blockIdx.x;               /* 1024 blocks: b(4) x R(16) x P(16) */
  int b  = blk >> 8;
  int R0 = ((blk >> 4) & 15) << 6;
  int P0 = (blk & 15) << 6;
  const float* semb = ws + OFF_SEMN + (size_t)b * SEM_C * N_;
  int tid  = threadIdx.x;
  int lane = tid & 31;
  int wv   = tid >> 5;                /* 0..7 */
  int m    = lane & 15;
  int half = lane >> 4;
  int kb   = half << 1;               /* K offset within 4: 0 or 2 */
  int tr   = wv >> 1;                 /* row tile 0..3 */
  int tc0  = (wv & 1) << 1;           /* col tiles {0,1} or {2,3} */
  int colA  = tr * 16 + m;
  int colB0 = tc0 * 16 + m;
  int colB1 = colB0 + 16;
  v8f acc0 = {}, acc1 = {};
  for (int c0 = 0; c0 < SEM_C; c0 += KC) {
    __syncthreads();
#ifdef USE_ASYNC_LDS
    for (int idx = tid; idx < KC * 16; idx += 256) {   /* 16 quads per row */
      int kk = idx >> 4, q = (idx & 15) << 2;
      __builtin_amdgcn_global_load_async_to_lds_b128(
          (as1_v4i)(semb + (size_t)(c0 + kk) * N_ + R0 + q),
          (as3_v4i)&As[kk * APAD + q], 0, 0);
      __builtin_amdgcn_global_load_async_to_lds_b128(
          (as1_v4i)(semb + (size_t)(c0 + kk) * N_ + P0 + q),
          (as3_v4i)&Bs[kk * APAD + q], 0, 0);
    }
    __builtin_amdgcn_s_wait_asynccnt(0);
#else
    for (int idx = tid; idx < KC * 16; idx += 256) {
      int kk = idx >> 4, q = (idx & 15) << 2;
      float4 ga = *(const float4*)(semb + (size_t)(c0 + kk) * N_ + R0 + q);
      float4 gb = *(const float4*)(semb + (size_t)(c0 + kk) * N_ + P0 + q);
      *(float4*)&As[kk * APAD + q] = ga;
      *(float4*)&Bs[kk * APAD + q] = gb;
    }
#endif
    __syncthreads();
#pragma unroll 4
    for (int k4 = 0; k4 < KC; k4 += 4) {
      int base = (k4 + kb) * APAD;
      v2f a, b0v, b1v;
      a.x   = As[base + colA];
      a.y   = As[base + APAD + colA];
      b0v.x = Bs[base + colB0];
      b0v.y = Bs[base + APAD + colB0];
      b1v.x = Bs[base + colB1];
      b1v.y = Bs[base + APAD + colB1];
      acc0 = __builtin_amdgcn_wmma_f32_16x16x4_f32(false, a, false, b0v,
                                                   (short)0, acc0, false, false);
      acc1 = __builtin_amdgcn_wmma_f32_16x16x4_f32(false, a, false, b1v,
                                                   (short)0, acc1, false, false);
    }
  }
  /* epilogue: wave's two tiles span exactly one 32-column block */
  int kblk = (P0 >> 5) + (tc0 >> 1);
  int rowb = R0 + tr * 16;
#pragma unroll
  for (int v = 0; v < 8; ++v) {
    float s0 = fmaxf(acc0[v], 0.0f);
    float s1 = fmaxf(acc1[v], 0.0f);
    float mx = fmaxf(s0, s1);
    float s  = s0 + s1;
#pragma unroll
    for (int off = 8; off; off >>= 1) {          /* reduce across 16 cols */
      s  += __shfl_xor(s, off, 32);
      mx  = fmaxf(mx, __shfl_xor(mx, off, 32));
    }
    if (m == 0) {
      int row = rowb + v + (half << 3);
      atomicAdd(&ws[OFF_S + ((size_t)b * N_ + row) * 32 + kblk], s);
      atomicMax((int*)&ws[OFF_MAX + b * N_ + row], __float_as_int(mx));
    }
  }
}

/* ---------------- t = S / (rowmax + 1e-5) ---------------- */
__global__ void t_finalize(float* __restrict__ ws, const int* __restrict__ flag) {
  if (*flag == 0) return;
  int i = blockIdx.x * 256 + threadIdx.x;
  if (i >= B_ * N_ * 32) return;
  ws[OFF_T + i] = ws[OFF_S + i] / (ws[OFF_MAX + (i >> 5)] + 1e-5f);
}

/* ---------------- sum_{r,q} |fo[r]-fo[q]| * t[r, q&31] ---------------- */
__global__ void at_contract(float* __restrict__ ws, const int* __restrict__ flag) {
  if (*flag == 0) return;
  __shared__ float fo_l[1024];
  __shared__ float t_l[256 * 33];
  __shared__ float red[8];
  int blk = blockIdx.x;          /* 0..335 */
  int rb = blk & 3;
  int c  = (blk >> 2) % C_OUT;
  int b  = blk / (C_OUT * 4);
  int tid = threadIdx.x;
  const float* fop = ws + OFF_FO + ((size_t)(b * C_OUT + c)) * N_;
  for (int j = tid; j < 1024; j += 256) fo_l[j] = fop[j];
  const float* tp = ws + OFF_T + ((size_t)b * N_ + rb * 256) * 32;
  for (int j = tid; j < 256 * 32; j += 256)
    t_l[(j >> 5) * 33 + (j & 31)] = tp[j];
  __syncthreads();
  int r = rb * 256 + tid;
  float xr = fo_l[r];
  const float* tr = &t_l[tid * 33];
  float acc = 0.0f;
#pragma unroll 4
  for (int q = 0; q < 1024; ++q)
    acc += fabsf(xr - fo_l[q]) * tr[q & 31];
  acc = wave_sum(acc);
  int lane = tid & 31, wid = tid >> 5;
  if (lane == 0) red[wid] = acc;
  __syncthreads();
  if (tid == 0) {
    float s = 0.f;
#pragma unroll
    for (int w = 0; w < 8; ++w) s += red[w];
    atomicAdd(&ws[OFF_TV], s);
  }
}

/* ---------------- fallback 2D gradient loss (flag==0 path) ---------------- */
__global__ void grad2d(const float* __restrict__ outp, float* __restrict__ ws,
                       const int* __restrict__ flag) {
  if (*flag != 0) return;
  __shared__ float red[8];
  const long NH = (long)B_ * C_OUT * 511 * 512;
  const long NW = (long)B_ * C_OUT * 512 * 511;
  long i = (long)blockIdx.x * 256 + threadIdx.x;
  long stride = (long)gridDim.x * 256;
  float acc = 0.f;
  for (long idx = i; idx < NH + NW; idx += stride) {
    if (idx < NH) {
      long w = idx % 512; long rest = idx / 512;
      long h = rest % 511; long bc = rest / 511;
      const float* p = outp + bc * (long)HW_ + h * 512 + w;
      acc += fabsf(p[512] - p[0]);
    } else {
      long j = idx - NH;
      long w = j % 511; long rest = j / 511;
      long h = rest % 512; long bc = rest / 512;
      const float* p = outp + bc * (long)HW_ + h * 512 + w;
      acc += fabsf(p[1] - p[0]);
    }
  }
  acc = wave_sum(acc);
  int lane = threadIdx.x & 31, wid = threadIdx.x >> 5;
  if (lane == 0) red[wid] = acc;
  __syncthreads();
  if (threadIdx.x == 0) {
    float s = 0.f;
#pragma unroll
    for (int w = 0; w < 8; ++w) s += red[w];
    atomicAdd(&ws[OFF_GRAD], s);
  }
}

/* ---------------- combine: loss_LS = sum(sq - num^2/den)/B ---------------- */
__global__ void finalize(float* __restrict__ ws, const int* __restrict__ flag,
                         float* __restrict__ outv) {
  __shared__ float red[8];
  int tid = threadIdx.x;
  float acc = 0.f;
  for (int i = tid; i < B_ * T_ * C_OUT; i += 256) {
    int b = i / (T_ * C_OUT);
    int c = i % C_OUT;
    float num = ws[OFF_NUM + i];
    float sq  = ws[OFF_SQ + i];
    float den = ws[OFF_DEN + b * C_OUT + c];
    acc += sq - num * num / den;
  }
  acc = wave_sum(acc);
  int lane = tid & 31, wid = tid >> 5;
  if (lane == 0) red[wid] = acc;
  __syncthreads();
  if (tid == 0) {
    float tot = 0.f;
#pragma unroll
    for (int w = 0; w < 8; ++w) tot += red[w];
    float ls  = tot / (float)B_;
    float raw = (*flag != 0) ? ws[OFF_TV] : ws[OFF_GRAD];
    float tv  = (raw / (float)B_) * LAMBDA_TV_F;
    outv[0] = ls * BETA_F;
    outv[1] = tv * BETA_F;
    outv[2] = (ls + tv) * BETA_F;
  }
}

extern "C" void kernel_launch(void* const* d_in, const int* in_sizes, int n_in,
                              void* d_out, int out_size, void* d_ws, size_t ws_size,
                              hipStream_t stream) {
  (void)in_sizes; (void)n_in; (void)out_size; (void)ws_size;
  const float* output   = (const float*)d_in[0];
  const float* target   = (const float*)d_in[1];
  const float* f_sem    = (const float*)d_in[2];
  const float* f_output = (const float*)d_in[3];
  const int*   flag     = (const int*)d_in[4];
  float* ws   = (float*)d_ws;
  float* outv = (float*)d_out;

  zero_accum<<<(ZERO_CNT + 255) / 256, 256, 0, stream>>>(ws);
  levelset_partial<<<dim3(128, 4), 256, 0, stream>>>(output, target, ws);
  resize_fo<<<(B_ * C_OUT * N_ + 255) / 256, 256, 0, stream>>>(f_output, ws, flag);
  sem_normalize<<<(B_ * N_ + 255) / 256, 256, 0, stream>>>(f_sem, ws, flag);
  tau_wmma<<<1024, 256, 0, stream>>>(ws, flag);   /* 64x64 macro-tiles, LDS-staged */
  t_finalize<<<(B_ * N_ * 32 + 255) / 256, 256, 0, stream>>>(ws, flag);
  at_contract<<<B_ * C_OUT * 4, 256, 0, stream>>>(ws, flag);
  grad2d<<<2048, 256, 0, stream>>>(output, ws, flag);
  finalize<<<1, 256, 0, stream>>>(ws, flag, outv);
}